// GCN_24618752541269
// MI455X (gfx1250) — compile-verified
//
#include <hip/hip_runtime.h>
#include <hip/hip_bf16.h>
#include <math.h>

// ---------------------------------------------------------------------------
// GCN forward:  out = log_softmax( A (relu( A (x@W1) + b1 )) @ W2 + b2 )
// A = sparse adjacency as (src, dst, weight) edge list, segment-sum by dst.
// fp32 WMMA (V_WMMA_F32_16X16X4_F32) for both dense GEMMs; atomic scatter SpMM.
// relu is fused into the second GEMM's A-fragment loads (saves a 51MB pass).
// W1/W2 are pre-packed into the WMMA B-fragment layout so the GEMM inner loop
// does one coalesced b64 load per fragment instead of two strided b32 loads.
// ---------------------------------------------------------------------------

typedef float v2f __attribute__((ext_vector_type(2)));
typedef float v8f __attribute__((ext_vector_type(8)));

#define NFEAT 256
#define NHID  64
#define NOUT  16

// ---------------------------------------------------------------------------
// Pre-pack W[K x N] (row major) into B-fragment order:
//   PB[kstep][tile][lane] = { W[4*kstep + 2*(lane>>4)    ][tile*16 + lane&15],
//                             W[4*kstep + 2*(lane>>4) + 1][tile*16 + lane&15] }
// One-time cost; W1 = 64KB, W2 = 4KB.
// ---------------------------------------------------------------------------
template <int K, int N>
__global__ void pack_b(const float* __restrict__ W, v2f* __restrict__ PB) {
    constexpr int NT    = N / 16;
    constexpr int TOTAL = (K / 4) * NT * 32;
    int idx = blockIdx.x * blockDim.x + threadIdx.x;
    if (idx >= TOTAL) return;
    const int lane  = idx & 31;
    const int tile  = (idx >> 5) % NT;
    const int kstep = idx / (NT * 32);
    const int half  = lane >> 4;
    const int col   = tile * 16 + (lane & 15);
    const int k     = kstep * 4 + 2 * half;
    v2f b;
    b.x = W[(size_t)k * N + col];
    b.y = W[(size_t)(k + 1) * N + col];
    PB[idx] = b;
}

// ---------------------------------------------------------------------------
// Dense GEMM: out[M x N] = act(A[M x K]) @ W[K x N], act = relu if RELU.
// One wave computes a 16 x N strip with V_WMMA_F32_16X16X4_F32.
//
// A-fragment (32-bit A 16x4, ISA 7.12.2): lanes 0-15 row=lane, K pair {k,k+1};
//   lanes 16-31 row=lane-16, K pair {k+2,k+3} -> contiguous float2 per lane.
// B-fragment comes pre-packed (see pack_b).
// C/D (16x16 f32): VGPR j -> row j (lanes 0-15) / j+8 (lanes 16-31); lane%16=col.
// ---------------------------------------------------------------------------
template <int K, int N, bool RELU>
__global__ __launch_bounds__(64)
void gemm_wmma_f32(const float* __restrict__ A,
                   const v2f* __restrict__ PB,
                   float* __restrict__ out, int M) {
    constexpr int NT = N / 16;
    const int wave  = (blockIdx.x * blockDim.x + threadIdx.x) >> 5;
    const int lane  = threadIdx.x & 31;
    const int mbase = wave * 16;
    if (mbase >= M) return;                 // wave-uniform: whole wave exits

    const int half = lane >> 4;
    const int l16  = lane & 15;

    int arow = mbase + l16;
    if (arow >= M) arow = M - 1;            // clamp tail loads (stores guarded)
    const float* __restrict__ ap = A + (size_t)arow * K + 2 * half;
    const v2f*   __restrict__ bp = PB + lane;

    v8f acc[NT];
#pragma unroll
    for (int t = 0; t < NT; ++t) acc[t] = (v8f){0.f,0.f,0.f,0.f,0.f,0.f,0.f,0.f};

#pragma unroll 8
    for (int k4 = 0; k4 < K / 4; ++k4) {
        v2f a = *(const v2f*)(ap + k4 * 4);
        if (RELU) { a.x = fmaxf(a.x, 0.f); a.y = fmaxf(a.y, 0.f); }
        const v2f* __restrict__ bk = bp + (size_t)k4 * (NT * 32);
#pragma unroll
        for (int t = 0; t < NT; ++t) {
            acc[t] = __builtin_amdgcn_wmma_f32_16x16x4_f32(
                false, a, false, bk[t * 32], (short)0, acc[t], false, false);
        }
    }

#pragma unroll
    for (int t = 0; t < NT; ++t) {
#pragma unroll
        for (int j = 0; j < 8; ++j) {
            const int row = mbase + j + 8 * half;
            if (row < M)
                out[(size_t)row * N + t * 16 + l16] = acc[t][j];
        }
    }
}

// ---------------------------------------------------------------------------
// acc[i][j] = bias[j]   (fold +b into the segment-sum accumulator init)
// ---------------------------------------------------------------------------
template <int D>
__global__ void init_bias(float* __restrict__ acc, const float* __restrict__ bias,
                          long long total) {
    long long i = (long long)blockIdx.x * blockDim.x + threadIdx.x;
    if (i < total) acc[i] = bias[i & (D - 1)];
}

// ---------------------------------------------------------------------------
// SpMM scatter: acc[dst[e], :] += w[e] * S[src[e], :]
// D/4 lanes per edge, float4 gather + 4 f32 atomics (global_atomic_add_f32).
// ---------------------------------------------------------------------------
template <int D>
__global__ void spmm_scatter(const float* __restrict__ S,
                             const int* __restrict__ src,
                             const int* __restrict__ dst,
                             const float* __restrict__ w,
                             float* __restrict__ acc, int n_edges) {
    constexpr int C = D / 4;                 // float4 chunks per edge
    long long t = (long long)blockIdx.x * blockDim.x + threadIdx.x;
    int e = (int)(t / C);
    int c = (int)(t % C);
    if (e >= n_edges) return;

    const int   s  = src[e];
    const int   d  = dst[e];
    const float wt = w[e];

    const float4 v = *((const float4*)(S + (size_t)s * D) + c);
    float* ap = acc + (size_t)d * D + c * 4;
    atomicAdd(ap + 0, wt * v.x);
    atomicAdd(ap + 1, wt * v.y);
    atomicAdd(ap + 2, wt * v.z);
    atomicAdd(ap + 3, wt * v.w);
}

// ---------------------------------------------------------------------------
// Row-wise log_softmax over NOUT=16 classes; one thread per node.
// ---------------------------------------------------------------------------
__global__ void logsoftmax16(const float* __restrict__ logits,
                             float* __restrict__ out, int n) {
    int i = blockIdx.x * blockDim.x + threadIdx.x;
    if (i >= n) return;

    float v[NOUT];
    const float4* p = (const float4*)(logits + (size_t)i * NOUT);
#pragma unroll
    for (int q = 0; q < NOUT / 4; ++q) {
        float4 t = p[q];
        v[q * 4 + 0] = t.x; v[q * 4 + 1] = t.y;
        v[q * 4 + 2] = t.z; v[q * 4 + 3] = t.w;
    }
    float mx = v[0];
#pragma unroll
    for (int j = 1; j < NOUT; ++j) mx = fmaxf(mx, v[j]);
    float s = 0.f;
#pragma unroll
    for (int j = 0; j < NOUT; ++j) s += expf(v[j] - mx);
    const float lse = mx + logf(s);

    float4* o = (float4*)(out + (size_t)i * NOUT);
#pragma unroll
    for (int q = 0; q < NOUT / 4; ++q) {
        float4 t;
        t.x = v[q * 4 + 0] - lse; t.y = v[q * 4 + 1] - lse;
        t.z = v[q * 4 + 2] - lse; t.w = v[q * 4 + 3] - lse;
        o[q] = t;
    }
}

// ---------------------------------------------------------------------------
extern "C" void kernel_launch(void* const* d_in, const int* in_sizes, int n_in,
                              void* d_out, int out_size, void* d_ws, size_t ws_size,
                              hipStream_t stream) {
    const float* x    = (const float*)d_in[0];
    const int*   esrc = (const int*)  d_in[1];
    const int*   edst = (const int*)  d_in[2];
    const float* ew   = (const float*)d_in[3];
    const float* W1   = (const float*)d_in[4];
    const float* b1   = (const float*)d_in[5];
    const float* W2   = (const float*)d_in[6];
    const float* b2   = (const float*)d_in[7];
    float* out = (float*)d_out;

    const int n_nodes = in_sizes[0] / NFEAT;   // 100000
    const int n_edges = in_sizes[1];           // 1600000

    // Packed-B fragment counts
    constexpr int PB1_N = (NFEAT / 4) * (NHID / 16) * 32;   // 8192 float2
    constexpr int PB2_N = (NHID  / 4) * (NOUT / 16) * 32;   //  512 float2

    // Workspace layout (floats): S1 | H | S2 | A2 | PB1 | PB2
    float* S1  = (float*)d_ws;                         // n_nodes * NHID
    float* H   = S1  + (size_t)n_nodes * NHID;         // n_nodes * NHID
    float* S2  = H   + (size_t)n_nodes * NHID;         // n_nodes * NOUT
    float* A2  = S2  + (size_t)n_nodes * NOUT;         // n_nodes * NOUT
    v2f*   PB1 = (v2f*)(A2 + (size_t)n_nodes * NOUT);  // PB1_N float2
    v2f*   PB2 = PB1 + PB1_N;                          // PB2_N float2

    const int mtiles      = (n_nodes + 15) / 16;       // waves per GEMM
    const int gemm_blocks = (mtiles + 1) / 2;          // 2 waves / 64-thr block

    // 0) Pack W1/W2 into WMMA B-fragment layout (one coalesced b64 per frag)
    pack_b<NFEAT, NHID><<<(PB1_N + 255) / 256, 256, 0, stream>>>(W1, PB1);
    pack_b<NHID,  NOUT><<<(PB2_N + 255) / 256, 256, 0, stream>>>(W2, PB2);

    // 1) S1 = x @ W1                        (fp32 WMMA)
    gemm_wmma_f32<NFEAT, NHID, false>
        <<<gemm_blocks, 64, 0, stream>>>(x, PB1, S1, n_nodes);

    // 2) H = b1 (broadcast); H += scatter(w * S1[src])
    {
        long long tot = (long long)n_nodes * NHID;
        init_bias<NHID><<<(int)((tot + 255) / 256), 256, 0, stream>>>(H, b1, tot);
        long long thr = (long long)n_edges * (NHID / 4);
        spmm_scatter<NHID><<<(int)((thr + 255) / 256), 256, 0, stream>>>(
            S1, esrc, edst, ew, H, n_edges);
    }

    // 3) S2 = relu(H) @ W2                  (fp32 WMMA, relu fused into A loads)
    gemm_wmma_f32<NHID, NOUT, true>
        <<<gemm_blocks, 64, 0, stream>>>(H, PB2, S2, n_nodes);

    // 4) A2 = b2 (broadcast); A2 += scatter(w * S2[src])
    {
        long long tot = (long long)n_nodes * NOUT;
        init_bias<NOUT><<<(int)((tot + 255) / 256), 256, 0, stream>>>(A2, b2, tot);
        long long thr = (long long)n_edges * (NOUT / 4);
        spmm_scatter<NOUT><<<(int)((thr + 255) / 256), 256, 0, stream>>>(
            S2, esrc, edst, ew, A2, n_edges);
    }

    // 5) out = log_softmax(A2, axis=1)
    logsoftmax16<<<(n_nodes + 255) / 256, 256, 0, stream>>>(A2, out, n_nodes);
}